// GATv2Block_395136991822
// MI455X (gfx1250) — compile-verified
//
#include <hip/hip_runtime.h>

#define N_NODES 50000
#define N_EDGES 800000
#define IN_DIM 128
#define OUT_DIM 128
#define HEADS 4
#define HEAD_DIM 32
#define EDGE_DIM 10
#define NEG_SLOPE 0.2f
#define LN_EPS 1e-5f

typedef __attribute__((ext_vector_type(2))) float v2f;
typedef __attribute__((ext_vector_type(8))) float v8f;

// ---------------------------------------------------------------------------
// Kernel 1: xl = x @ w_l^T + b_l ; xr = x @ w_r^T + b_r   (fp32 WMMA 16x16x4)
// One wave computes one 16(node) x 16(outdim) tile for BOTH weight matrices,
// sharing the A operand. Block = 256 threads = 8 waves = full 128 out dim.
// Grid = N/16 = 3125 blocks (N divisible by 16).
// ---------------------------------------------------------------------------
__global__ __launch_bounds__(256) void gatv2_gemm_lr(
    const float* __restrict__ x,
    const float* __restrict__ wl, const float* __restrict__ bl,
    const float* __restrict__ wr, const float* __restrict__ br,
    float* __restrict__ xl, float* __restrict__ xr)
{
    const int wave = threadIdx.x >> 5;
    const int lane = threadIdx.x & 31;
    const int mbase = blockIdx.x * 16;   // node tile
    const int nbase = wave * 16;         // out-channel tile
    const int lm = lane & 15;
    const int kh = (lane >> 4) * 2;      // K sub-offset within 4-wide step

    const int arow = mbase + lm;         // node row for A operand
    const int bcol = nbase + lm;         // out channel for B operand

    const float* __restrict__ xrow  = x  + (size_t)arow * IN_DIM;
    const float* __restrict__ wlrow = wl + (size_t)bcol * IN_DIM;
    const float* __restrict__ wrrow = wr + (size_t)bcol * IN_DIM;

    v8f accL = {};
    v8f accR = {};
    #pragma unroll 4
    for (int k = 0; k < IN_DIM; k += 4) {
        v2f a, bL, bR;
        a.x  = xrow[k + kh];      a.y  = xrow[k + kh + 1];
        bL.x = wlrow[k + kh];     bL.y = wlrow[k + kh + 1];
        bR.x = wrrow[k + kh];     bR.y = wrrow[k + kh + 1];
        accL = __builtin_amdgcn_wmma_f32_16x16x4_f32(
            false, a, false, bL, (short)0, accL, false, false);
        accR = __builtin_amdgcn_wmma_f32_16x16x4_f32(
            false, a, false, bR, (short)0, accR, false, false);
    }

    const float biasL = bl[bcol];
    const float biasR = br[bcol];
    const int rbase = mbase + (lane >> 4) * 8;   // D: lanes 16-31 hold M=8..15
    #pragma unroll
    for (int r = 0; r < 8; ++r) {
        const size_t off = (size_t)(rbase + r) * OUT_DIM + bcol;
        xl[off] = accL[r] + biasL;
        xr[off] = accR[r] + biasR;
    }
}

// ---------------------------------------------------------------------------
// Kernel 2: init scratch (must run every launch; harness does not re-poison)
// ---------------------------------------------------------------------------
__global__ void gatv2_init(float* __restrict__ maxv, float* __restrict__ denom,
                           float* __restrict__ outacc)
{
    const int i = blockIdx.x * blockDim.x + threadIdx.x;
    if (i < N_NODES * HEADS) {
        maxv[i]  = -__builtin_inff();
        denom[i] = 0.0f;
    }
    if (i < N_NODES * OUT_DIM) outacc[i] = 0.0f;
}

// ---------------------------------------------------------------------------
// Kernel 3: per-edge attention logits + segment max.
// One wave per edge; lane = channel within head; ea computed on the fly
// (w_e fits trivially in L2; never materialize the 400 MB ea tensor).
// ---------------------------------------------------------------------------
__global__ __launch_bounds__(256) void gatv2_edge_logits(
    const int* __restrict__ ei, const float* __restrict__ eattr,
    const float* __restrict__ we, const float* __restrict__ att,
    const float* __restrict__ xl, const float* __restrict__ xr,
    float* __restrict__ logits, float* __restrict__ maxv)
{
    const int e = blockIdx.x * 8 + (threadIdx.x >> 5);
    if (e >= N_EDGES) return;
    const int lane = threadIdx.x & 31;
    const int src = ei[e];
    const int dst = ei[N_EDGES + e];

    float ea[EDGE_DIM];
    #pragma unroll
    for (int d = 0; d < EDGE_DIM; ++d)
        ea[d] = eattr[(size_t)e * EDGE_DIM + d];

    #pragma unroll
    for (int h = 0; h < HEADS; ++h) {
        const int c = h * HEAD_DIM + lane;
        float ev = 0.0f;
        #pragma unroll
        for (int d = 0; d < EDGE_DIM; ++d)
            ev = fmaf(ea[d], we[(size_t)c * EDGE_DIM + d], ev);
        float v = xl[(size_t)src * OUT_DIM + c] +
                  xr[(size_t)dst * OUT_DIM + c] + ev;
        v = v > 0.0f ? v : NEG_SLOPE * v;
        float p = v * att[c];
        #pragma unroll
        for (int off = 16; off > 0; off >>= 1)
            p += __shfl_xor(p, off, 32);
        if (lane == 0) {
            logits[(size_t)e * HEADS + h] = p;
            atomicMax(&maxv[(size_t)dst * HEADS + h], p);
        }
    }
}

// ---------------------------------------------------------------------------
// Kernel 4: ex = exp(logit - max[dst]); accumulate softmax denominator.
// ---------------------------------------------------------------------------
__global__ void gatv2_edge_expsum(
    const int* __restrict__ ei, float* __restrict__ logits,
    const float* __restrict__ maxv, float* __restrict__ denom)
{
    const int i = blockIdx.x * blockDim.x + threadIdx.x;
    if (i >= N_EDGES * HEADS) return;
    const int e = i >> 2;
    const int h = i & 3;
    const int dst = ei[N_EDGES + e];
    const float ex = __expf(logits[i] - maxv[(size_t)dst * HEADS + h]);
    logits[i] = ex;           // overwrite with exp value
    atomicAdd(&denom[(size_t)dst * HEADS + h], ex);
}

// ---------------------------------------------------------------------------
// Kernel 5: alpha-weighted scatter-add aggregation. One wave per edge,
// lane = channel within head, 128 fp32 atomics per edge.
// ---------------------------------------------------------------------------
__global__ __launch_bounds__(256) void gatv2_edge_aggregate(
    const int* __restrict__ ei, const float* __restrict__ logits,
    const float* __restrict__ denom, const float* __restrict__ xl,
    float* __restrict__ outacc)
{
    const int e = blockIdx.x * 8 + (threadIdx.x >> 5);
    if (e >= N_EDGES) return;
    const int lane = threadIdx.x & 31;
    const int src = ei[e];
    const int dst = ei[N_EDGES + e];
    #pragma unroll
    for (int h = 0; h < HEADS; ++h) {
        const float alpha = logits[(size_t)e * HEADS + h] /
                            (denom[(size_t)dst * HEADS + h] + 1e-16f);
        const int c = h * HEAD_DIM + lane;
        atomicAdd(&outacc[(size_t)dst * OUT_DIM + c],
                  xl[(size_t)src * OUT_DIM + c] * alpha);
    }
}

// ---------------------------------------------------------------------------
// Kernel 6: bias -> SiLU -> residual -> LayerNorm. One wave per node,
// each lane owns 4 channels; mean/var via wave32 shfl_xor reduction.
// ---------------------------------------------------------------------------
__global__ __launch_bounds__(256) void gatv2_final(
    const float* __restrict__ x, const float* __restrict__ outacc,
    const float* __restrict__ cbias, const float* __restrict__ gamma,
    const float* __restrict__ beta, float* __restrict__ out)
{
    const int n = blockIdx.x * 8 + (threadIdx.x >> 5);
    if (n >= N_NODES) return;
    const int lane = threadIdx.x & 31;

    float h[4];
    float s = 0.0f;
    #pragma unroll
    for (int j = 0; j < 4; ++j) {
        const int c = lane + 32 * j;
        const float v = outacc[(size_t)n * OUT_DIM + c] + cbias[c];
        const float sig = 1.0f / (1.0f + __expf(-v));
        h[j] = v * sig + x[(size_t)n * OUT_DIM + c];
        s += h[j];
    }
    #pragma unroll
    for (int off = 16; off > 0; off >>= 1) s += __shfl_xor(s, off, 32);
    const float mu = s * (1.0f / OUT_DIM);

    float vs = 0.0f;
    #pragma unroll
    for (int j = 0; j < 4; ++j) { const float d = h[j] - mu; vs += d * d; }
    #pragma unroll
    for (int off = 16; off > 0; off >>= 1) vs += __shfl_xor(vs, off, 32);
    const float inv = rsqrtf(vs * (1.0f / OUT_DIM) + LN_EPS);

    #pragma unroll
    for (int j = 0; j < 4; ++j) {
        const int c = lane + 32 * j;
        out[(size_t)n * OUT_DIM + c] = (h[j] - mu) * inv * gamma[c] + beta[c];
    }
}

// ---------------------------------------------------------------------------
extern "C" void kernel_launch(void* const* d_in, const int* in_sizes, int n_in,
                              void* d_out, int out_size, void* d_ws, size_t ws_size,
                              hipStream_t stream) {
    const float* x     = (const float*)d_in[0];
    const int*   ei    = (const int*)  d_in[1];
    const float* eattr = (const float*)d_in[2];
    const float* wl    = (const float*)d_in[3];
    const float* bl    = (const float*)d_in[4];
    const float* wr    = (const float*)d_in[5];
    const float* br    = (const float*)d_in[6];
    const float* we    = (const float*)d_in[7];
    const float* att   = (const float*)d_in[8];
    const float* cb    = (const float*)d_in[9];
    const float* gam   = (const float*)d_in[10];
    const float* bet   = (const float*)d_in[11];
    float* out = (float*)d_out;

    float* ws     = (float*)d_ws;
    float* xl     = ws;                                   // N*128
    float* xr     = xl     + (size_t)N_NODES * OUT_DIM;   // N*128
    float* outacc = xr     + (size_t)N_NODES * OUT_DIM;   // N*128
    float* logits = outacc + (size_t)N_NODES * OUT_DIM;   // E*4
    float* maxv   = logits + (size_t)N_EDGES * HEADS;     // N*4
    float* denom  = maxv   + (size_t)N_NODES * HEADS;     // N*4

    gatv2_gemm_lr<<<N_NODES / 16, 256, 0, stream>>>(x, wl, bl, wr, br, xl, xr);

    const int initN = N_NODES * OUT_DIM;
    gatv2_init<<<(initN + 255) / 256, 256, 0, stream>>>(maxv, denom, outacc);

    gatv2_edge_logits<<<(N_EDGES + 7) / 8, 256, 0, stream>>>(
        ei, eattr, we, att, xl, xr, logits, maxv);

    gatv2_edge_expsum<<<(N_EDGES * HEADS + 255) / 256, 256, 0, stream>>>(
        ei, logits, maxv, denom);

    gatv2_edge_aggregate<<<(N_EDGES + 7) / 8, 256, 0, stream>>>(
        ei, logits, denom, xl, outacc);

    gatv2_final<<<(N_NODES + 7) / 8, 256, 0, stream>>>(
        x, outacc, cb, gam, bet, out);
}